// Word2Vec_2345052144357
// MI455X (gfx1250) — compile-verified
//
#include <hip/hip_runtime.h>
#include <hip/hip_bf16.h>

#define VOC    50257
#define EMB    200
#define NROWS  4096
#define KSTEPS (EMB / 4)     // 50 WMMA K-steps of 4
#define CT     8             // column tiles (of 16) per workgroup -> 128 cols
#define NWAVES 16            // 512 threads
#define RT     (NROWS / 16)  // 256 row tiles

typedef __attribute__((ext_vector_type(2))) float v2f;
typedef __attribute__((ext_vector_type(8))) float v8f;

// ---------------------------------------------------------------------------
// Kernel 1: embedding gather.  emb[n][e] = A_w[e][x[n]] + A_b[e]
// ---------------------------------------------------------------------------
__global__ void emb_gather_kernel(const int* __restrict__ x,
                                  const float* __restrict__ A_w,
                                  const float* __restrict__ A_b,
                                  float* __restrict__ emb) {
    const int n = blockIdx.x;
    const int e = threadIdx.x;
    if (e < EMB) {
        const int xn = x[n];
        emb[(size_t)n * EMB + e] = A_w[(size_t)e * VOC + xn] + A_b[e];
    }
}

// ---------------------------------------------------------------------------
// One 16x16 logits tile: A fragments cached in registers (a[50]),
// B fragments streamed from LDS (one 8B ds_load per WMMA, conflict-free).
// bBase must derive from an opaque value so LICM cannot hoist the LDS loads
// out of the row-tile loop (that hoist spilled 800 VGPRs to scratch in v2).
// ---------------------------------------------------------------------------
__device__ __forceinline__ v8f compute_tile(const v2f* __restrict__ a,
                                            const float* __restrict__ bBase) {
    v8f acc0 = {};
    v8f acc1 = {};
#pragma unroll
    for (int step = 0; step < KSTEPS; step += 2) {
        v2f b0 = *(const v2f*)(bBase + 64 * step);
        acc0 = __builtin_amdgcn_wmma_f32_16x16x4_f32(
            false, a[step], false, b0, (short)0, acc0, false, false);
        v2f b1 = *(const v2f*)(bBase + 64 * step + 64);
        acc1 = __builtin_amdgcn_wmma_f32_16x16x4_f32(
            false, a[step + 1], false, b1, (short)0, acc1, false, false);
    }
    return acc0 + acc1;
}

// Opaque zero: regenerated inside each loop iteration so LDS addresses are
// not provably loop-invariant (defeats LICM without making loads volatile).
__device__ __forceinline__ int opaque_zero() {
    int z;
    asm volatile("s_mov_b32 %0, 0" : "=s"(z));
    return z;
}

// ---------------------------------------------------------------------------
// Kernel 2: fused GEMM + column softmax, 128 columns per workgroup.
// 512 threads = 16 wave32s; each wave owns 16 row tiles, all 8 column tiles.
// A fragments loaded once per row tile, reused across the 8 column tiles.
// Sweep 1: online per-column max + rescaled sum.  Sweep 2: recompute, write
// exp(l - M)/S exactly once (store-bound by design).
// B_b skipped: softmax over axis 0 is invariant to per-column constants.
// ---------------------------------------------------------------------------
__global__ __launch_bounds__(512)
void logits_softmax_kernel(const float* __restrict__ emb,
                           const float* __restrict__ B_w,
                           float* __restrict__ out) {
    // B tile: element (coltile j, k, c) at j*3200 + (k>>1)*32 + 2*c + (k&1)
    __shared__ float ldsB[CT * 16 * EMB];        // 100 KB
    __shared__ float redM[NWAVES][CT * 16];      // 8 KB
    __shared__ float redS[NWAVES][CT * 16];      // 8 KB

    const int v0   = blockIdx.x * (CT * 16);
    const int tid  = threadIdx.x;
    const int lane = tid & 31;
    const int wave = tid >> 5;
    const int c    = lane & 15;   // column within tile (also A-row-in-tile)
    const int hg   = lane >> 4;   // half-group: K-pair select / D row offset

    // Stage B tile (coalesced 200-float row chunks from B_w, scatter to LDS)
    for (int i = tid; i < CT * 16 * EMB; i += NWAVES * 32) {
        const int cc = i / EMB;          // 0..127
        const int k  = i % EMB;
        int v = v0 + cc;
        if (v > VOC - 1) v = VOC - 1;    // clamp for ragged last block
        ldsB[(cc >> 4) * (16 * EMB) + (k >> 1) * 32 + (cc & 15) * 2 + (k & 1)]
            = B_w[(size_t)v * EMB + k];
    }
    __syncthreads();

    const int laneOff = hg * 32 + 2 * c;   // per-lane B offset within a coltile

    // ---- Sweep 1: online per-column max & rescaled sum ---------------------
    float m[CT], s[CT];
#pragma unroll
    for (int j = 0; j < CT; ++j) { m[j] = -3.402823466e38f; s[j] = 0.0f; }

    for (int t = wave; t < RT; t += NWAVES) {
        const float* aRow = emb + (size_t)(t * 16 + c) * EMB + 2 * hg;
        v2f a[KSTEPS];
#pragma unroll
        for (int i = 0; i < KSTEPS; ++i) a[i] = *(const v2f*)(aRow + 4 * i);

        const float* ldsBt = ldsB + opaque_zero() + laneOff;  // opaque base
#pragma unroll
        for (int j = 0; j < CT; ++j) {
            v8f acc = compute_tile(a, ldsBt + j * (16 * EMB));
            float tm = acc[0];
#pragma unroll
            for (int r = 1; r < 8; ++r) tm = fmaxf(tm, acc[r]);
            const float nm = fmaxf(m[j], tm);
            float ts = 0.0f;
#pragma unroll
            for (int r = 0; r < 8; ++r) ts += __expf(acc[r] - nm);
            s[j] = s[j] * __expf(m[j] - nm) + ts;
            m[j] = nm;
        }
    }

    // Combine lane <-> lane^16 (same column, other 8 rows of each tile)
#pragma unroll
    for (int j = 0; j < CT; ++j) {
        const float m2 = __shfl_xor(m[j], 16, 32);
        const float s2 = __shfl_xor(s[j], 16, 32);
        const float mm = fmaxf(m[j], m2);
        const float ss = s[j] * __expf(m[j] - mm) + s2 * __expf(m2 - mm);
        if (lane < 16) { redM[wave][j * 16 + lane] = mm;
                         redS[wave][j * 16 + lane] = ss; }
    }
    __syncthreads();

    // Combine the 16 waves (every thread redundantly, for its columns)
    float Mg[CT], Inv[CT];
#pragma unroll
    for (int j = 0; j < CT; ++j) {
        const int col = j * 16 + c;
        float mg = -3.402823466e38f;
#pragma unroll
        for (int w = 0; w < NWAVES; ++w) mg = fmaxf(mg, redM[w][col]);
        float sg = 0.0f;
#pragma unroll
        for (int w = 0; w < NWAVES; ++w)
            sg += redS[w][col] * __expf(redM[w][col] - mg);
        Mg[j]  = mg;
        Inv[j] = 1.0f / sg;
    }

    // ---- Sweep 2: recompute tiles, write normalized softmax once -----------
    for (int t = wave; t < RT; t += NWAVES) {
        const float* aRow = emb + (size_t)(t * 16 + c) * EMB + 2 * hg;
        v2f a[KSTEPS];
#pragma unroll
        for (int i = 0; i < KSTEPS; ++i) a[i] = *(const v2f*)(aRow + 4 * i);

        const float* ldsBt = ldsB + opaque_zero() + laneOff;  // opaque base
#pragma unroll
        for (int j = 0; j < CT; ++j) {
            v8f acc = compute_tile(a, ldsBt + j * (16 * EMB));
            const int vcol = v0 + j * 16 + c;
            if (vcol < VOC) {
#pragma unroll
                for (int r = 0; r < 8; ++r) {
                    const int row = t * 16 + r + 8 * hg;   // D: M = r + 8*hg
                    out[(size_t)row * VOC + vcol] = __expf(acc[r] - Mg[j]) * Inv[j];
                }
            }
        }
    }
}

// ---------------------------------------------------------------------------
// Inputs: x [N] i32, A_w [EMB*VOC] f32, A_b [EMB] f32, B_w [VOC*EMB] f32,
// B_b [VOC] f32 (unused: cancels in column softmax).  Output: [N*VOC] f32.
// Workspace: emb, NROWS*EMB*4 = 3.28 MB.
// ---------------------------------------------------------------------------
extern "C" void kernel_launch(void* const* d_in, const int* in_sizes, int n_in,
                              void* d_out, int out_size, void* d_ws, size_t ws_size,
                              hipStream_t stream) {
    (void)in_sizes; (void)n_in; (void)out_size; (void)ws_size;
    const int*   x   = (const int*)d_in[0];
    const float* A_w = (const float*)d_in[1];
    const float* A_b = (const float*)d_in[2];
    const float* B_w = (const float*)d_in[3];
    float* out = (float*)d_out;
    float* emb = (float*)d_ws;

    emb_gather_kernel<<<NROWS, 256, 0, stream>>>(x, A_w, A_b, emb);

    const int nblocks = (VOC + CT * 16 - 1) / (CT * 16);   // 393 column stripes
    logits_softmax_kernel<<<nblocks, 512, 0, stream>>>(emb, B_w, out);
}